// SequentialSSM_68564857913723
// MI455X (gfx1250) — compile-verified
//
#include <hip/hip_runtime.h>

// ---------------------------------------------------------------------------
// SequentialSSM (Mamba stack) for gfx1250 — fp32 WMMA (V_WMMA_F32_16X16X4_F32)
// B=2, D=256, L=512, NL=2 per direction, DS=16, DC=4
// Group a: d_model=512, seq=256, d_inner=1024, r=32
// Group b: d_model=256, seq=512, d_inner=512,  r=16
// ---------------------------------------------------------------------------

typedef __attribute__((ext_vector_type(2))) float v2f;
typedef __attribute__((ext_vector_type(8))) float v8f;

#define DS_N 16

__device__ __forceinline__ v8f wmma4(v2f a, v2f b, v8f c) {
    return __builtin_amdgcn_wmma_f32_16x16x4_f32(false, a, false, b,
                                                 (short)0, c, false, false);
}

// Inner K-loop, NT = number of valid 16-wide N subtiles (1 or 2).
// Branch-free; guards are resolved before entry so EXEC stays all-ones.
template <int NT>
__device__ __forceinline__ void gemm_core(
    const float* __restrict__ A0, const float* __restrict__ A1,
    const float* __restrict__ W0, const float* __restrict__ W1, int K,
    v8f& c00, v8f& c10, v8f& c01, v8f& c11)
{
#pragma unroll 4
    for (int k = 0; k < K; k += 4) {
        v2f a0 = *(const v2f*)(A0 + k);
        v2f a1 = *(const v2f*)(A1 + k);
        v2f b0 = *(const v2f*)(W0 + k);
        c00 = wmma4(a0, b0, c00);
        c10 = wmma4(a1, b0, c10);
        if constexpr (NT == 2) {
            v2f b1 = *(const v2f*)(W1 + k);
            c01 = wmma4(a0, b1, c01);
            c11 = wmma4(a1, b1, c11);
        }
    }
}

// ---------------------------------------------------------------------------
// C(M,N) = A(M,K; row stride lda) * W(N,K)^T, optional epilogue:
//   epi==1: C = softplus(C + bias[col])
// Block: 256 threads = 8 waves arranged 4(M) x 2(N); each wave owns 32x32 of C
// via four 16x16 WMMA accumulators. M % 128 == 0, K % 4 == 0 required.
// N-edge guards are scalarized via readfirstlane (wave-uniform by construction)
// so they compile to s_cbranch_scc, never EXEC manipulation around WMMA.
// ---------------------------------------------------------------------------
__global__ __launch_bounds__(256) void gemm_f32_wmma(
    const float* __restrict__ A, int lda,
    const float* __restrict__ W,          // (N,K) row-major
    float* __restrict__ C, int ldc,
    int M, int N, int K,
    const float* __restrict__ bias, int epi)
{
    const int tid  = threadIdx.x;
    const int wid  = tid >> 5;
    const int lane = tid & 31;
    const int l15  = lane & 15;
    const int kofs = (lane >> 4) << 1;   // lanes 0-15 -> K+0/1, lanes 16-31 -> K+2/3
    const int half8 = (lane >> 4) << 3;  // C rows: +8 for upper half-wave

    const int m0 = blockIdx.y * 128 + (wid >> 1) * 32;
    const int n0 = blockIdx.x * 64  + (wid & 1) * 32;

    // wave-uniform scalar guards (SGPR branches, EXEC untouched)
    const int n0ok = __builtin_amdgcn_readfirstlane((n0      < N) ? 1 : 0);
    const int n1ok = __builtin_amdgcn_readfirstlane((n0 + 16 < N) ? 1 : 0);
    if (!n0ok) return;   // whole wave out of range

    v8f c00 = {}, c01 = {}, c10 = {}, c11 = {};

    const float* A0 = A + (size_t)(m0 + l15)      * lda + kofs;
    const float* A1 = A + (size_t)(m0 + 16 + l15) * lda + kofs;
    const float* W0 = W + (size_t)(n0 + l15) * K + kofs;
    const float* W1 = W + (size_t)(n1ok ? (n0 + 16 + l15) : (n0 + l15)) * K + kofs;

    if (n1ok) gemm_core<2>(A0, A1, W0, W1, K, c00, c10, c01, c11);
    else      gemm_core<1>(A0, A1, W0, W1, K, c00, c10, c01, c11);

    auto store_tile = [&](const v8f& c, int mt, int nt) {
        const int col = n0 + nt + l15;
#pragma unroll
        for (int v = 0; v < 8; ++v) {
            const int row = m0 + mt + half8 + v;
            float val = c[v];
            if (epi == 1) {
                val += bias[col];
                val = (val > 20.f) ? val : __logf(1.f + __expf(val));   // softplus
            }
            C[(size_t)row * ldc + col] = val;
        }
    };
    store_tile(c00, 0, 0);
    store_tile(c10, 16, 0);
    if (n1ok) {
        store_tile(c01, 0, 16);
        store_tile(c11, 16, 16);
    }
}

// ---------------------------------------------------------------------------
// Causal depthwise conv (width 4, left pad 3) + bias + SiLU.
// Input: first-di half of xz (B,S,2*di); output xi (B,S,di).
// ---------------------------------------------------------------------------
__global__ void conv_silu(const float* __restrict__ xz,
                          const float* __restrict__ cw,   // (di,4)
                          const float* __restrict__ cb,   // (di)
                          float* __restrict__ xi,
                          int Bn, int S, int di)
{
    const int idx = blockIdx.x * blockDim.x + threadIdx.x;
    const int tot = Bn * S * di;
    if (idx >= tot) return;
    const int c = idx % di;
    const int s = (idx / di) % S;
    const int b = idx / (di * S);
    const int di2 = 2 * di;
    float acc = cb[c];
#pragma unroll
    for (int j = 0; j < 4; ++j) {
        const int ss = s - 3 + j;
        if (ss >= 0) acc += cw[c * 4 + j] * xz[((size_t)(b * S + ss)) * di2 + c];
    }
    xi[idx] = acc / (1.f + __expf(-acc));   // silu
}

// ---------------------------------------------------------------------------
// Selective scan. One thread per (b, channel); 16 states + 16 A values in
// registers; B(t)/C(t) staged through LDS (shared by all channels of a block).
// Fuses + D*xi and * silu(z). di % 256 == 0.
// ---------------------------------------------------------------------------
__global__ __launch_bounds__(256) void ssm_scan(
    const float* __restrict__ dt,    // (B,S,di)
    const float* __restrict__ xi,    // (B,S,di)
    const float* __restrict__ xdbl,  // (B,S,ldx); B at [r..r+16), C at [r+16..r+32)
    int ldx, int r,
    const float* __restrict__ xz,    // (B,S,2*di); z at +di
    const float* __restrict__ A_log, // (di,16)
    const float* __restrict__ Dp,    // (di)
    float* __restrict__ ys,          // (B,S,di)
    int S, int di)
{
    const int c = blockIdx.x * 256 + threadIdx.x;
    const int b = blockIdx.y;
    __shared__ float sB[DS_N];
    __shared__ float sC[DS_N];

    float Areg[DS_N], h[DS_N];
#pragma unroll
    for (int n = 0; n < DS_N; ++n) {
        Areg[n] = -__expf(A_log[(size_t)c * DS_N + n]);
        h[n] = 0.f;
    }
    const float dpc = Dp[c];
    const int di2 = 2 * di;

    for (int s = 0; s < S; ++s) {
        const size_t base = (size_t)(b * S + s);
        __syncthreads();
        if (threadIdx.x < 32) {
            const int n = threadIdx.x & 15;
            const float v = xdbl[base * ldx + r + (threadIdx.x < 16 ? 0 : DS_N) + n];
            if (threadIdx.x < 16) sB[n] = v; else sC[n] = v;
        }
        __syncthreads();

        const float dtv = dt[base * di + c];
        const float xiv = xi[base * di + c];
        const float zv  = xz[base * di2 + di + c];
        float y = 0.f;
#pragma unroll
        for (int n = 0; n < DS_N; ++n) {
            const float dA = __expf(dtv * Areg[n]);
            h[n] = dA * h[n] + dtv * sB[n] * xiv;
            y += h[n] * sC[n];
        }
        y = (y + dpc * xiv) * (zv / (1.f + __expf(-zv)));
        ys[base * di + c] = y;
    }
}

// ---------------------------------------------------------------------------
// Per-batch transpose: out[b, c, r] = in[b, r, c]. rows, cols % 32 == 0.
// ---------------------------------------------------------------------------
__global__ __launch_bounds__(256) void transpose_k(
    const float* __restrict__ in, float* __restrict__ out, int rows, int cols)
{
    __shared__ float tile[32][33];
    const int b = blockIdx.z;
    const float* ip = in  + (size_t)b * rows * cols;
    float*       op = out + (size_t)b * rows * cols;
    const int r0 = blockIdx.y * 32, c0 = blockIdx.x * 32;
    const int tx = threadIdx.x & 31, ty = threadIdx.x >> 5;
    for (int i = ty; i < 32; i += 8)
        tile[i][tx] = ip[(size_t)(r0 + i) * cols + (c0 + tx)];
    __syncthreads();
    for (int i = ty; i < 32; i += 8)
        op[(size_t)(c0 + i) * rows + (r0 + tx)] = tile[tx][i];
}

// ---------------------------------------------------------------------------
// out[b,d,l] = x[b,d,l] + h[b,l,d]   (final transpose-back + residual)
// ---------------------------------------------------------------------------
__global__ void residual_k(const float* __restrict__ x, const float* __restrict__ h,
                           float* __restrict__ out, int Bn, int Dd, int Ll)
{
    const int idx = blockIdx.x * blockDim.x + threadIdx.x;
    const int tot = Bn * Dd * Ll;
    if (idx >= tot) return;
    const int l = idx % Ll;
    const int d = (idx / Ll) % Dd;
    const int b = idx / (Ll * Dd);
    out[idx] = x[idx] + h[((size_t)b * Ll + l) * Dd + d];
}

// ---------------------------------------------------------------------------
// Host-side orchestration
// ---------------------------------------------------------------------------
struct MambaGroup {
    const float *in_w, *conv_w, *conv_b, *xproj_w, *dt_w, *dt_b, *A_log, *Dp, *out_w;
    int S, dm, di, r;
};

extern "C" void kernel_launch(void* const* d_in, const int* in_sizes, int n_in,
                              void* d_out, int out_size, void* d_ws, size_t ws_size,
                              hipStream_t stream)
{
    (void)in_sizes; (void)n_in; (void)out_size; (void)ws_size;
    const float* x = (const float*)d_in[0];

    MambaGroup ga{ (const float*)d_in[1], (const float*)d_in[2], (const float*)d_in[3],
                   (const float*)d_in[4], (const float*)d_in[5], (const float*)d_in[6],
                   (const float*)d_in[7], (const float*)d_in[8], (const float*)d_in[9],
                   /*S=*/256, /*dm=*/512, /*di=*/1024, /*r=*/32 };
    MambaGroup gb{ (const float*)d_in[10], (const float*)d_in[11], (const float*)d_in[12],
                   (const float*)d_in[13], (const float*)d_in[14], (const float*)d_in[15],
                   (const float*)d_in[16], (const float*)d_in[17], (const float*)d_in[18],
                   /*S=*/512, /*dm=*/256, /*di=*/512, /*r=*/16 };

    // workspace layout (floats)
    float* ws   = (float*)d_ws;
    float* xz   = ws;                  // max 2*256*2048 = 1,048,576
    float* xib  = xz   + 1048576;      // max 524,288
    float* xdbl = xib  + 524288;       // max 1024*48 = 49,152
    float* dtb  = xdbl + 49152;        // max 524,288
    float* ysb  = dtb  + 524288;       // max 524,288
    float* buf0 = ysb  + 524288;       // 262,144
    float* buf1 = buf0 + 262144;       // 262,144

    auto run_stack = [&](const MambaGroup& g, const float* input, float* outbuf) {
        const int Mrows = 2 * g.S;           // B * seq, multiple of 128
        const int ldx   = g.r + 2 * DS_N;    // x_dbl row stride
        for (int k = 0; k < 2; ++k) {
            const float* in = (k == 0) ? input : outbuf;

            // 1) xz = in @ in_w^T     (M x 2di)
            dim3 g1((unsigned)((2 * g.di + 63) / 64), (unsigned)(Mrows / 128));
            gemm_f32_wmma<<<g1, 256, 0, stream>>>(
                in, g.dm, g.in_w + (size_t)k * 2 * g.di * g.dm,
                xz, 2 * g.di, Mrows, 2 * g.di, g.dm, nullptr, 0);

            // 2) xi = silu(conv(xz[:, :di]) + conv_b)
            const int tot = Mrows * g.di;
            conv_silu<<<(tot + 255) / 256, 256, 0, stream>>>(
                xz, g.conv_w + (size_t)k * g.di * 4, g.conv_b + (size_t)k * g.di,
                xib, 2, g.S, g.di);

            // 3) x_dbl = xi @ xproj_w^T    (M x (r+32))
            dim3 g2((unsigned)((ldx + 63) / 64), (unsigned)(Mrows / 128));
            gemm_f32_wmma<<<g2, 256, 0, stream>>>(
                xib, g.di, g.xproj_w + (size_t)k * ldx * g.di,
                xdbl, ldx, Mrows, ldx, g.di, nullptr, 0);

            // 4) dt = softplus(x_dbl[:, :r] @ dt_w^T + dt_b)   (M x di)
            dim3 g3((unsigned)((g.di + 63) / 64), (unsigned)(Mrows / 128));
            gemm_f32_wmma<<<g3, 256, 0, stream>>>(
                xdbl, ldx, g.dt_w + (size_t)k * g.di * g.r,
                dtb, g.di, Mrows, g.di, g.r, g.dt_b + (size_t)k * g.di, 1);

            // 5) selective scan (+ D*xi, * silu(z))
            dim3 gs((unsigned)(g.di / 256), 2u);
            ssm_scan<<<gs, 256, 0, stream>>>(
                dtb, xib, xdbl, ldx, g.r, xz,
                g.A_log + (size_t)k * g.di * DS_N, g.Dp + (size_t)k * g.di,
                ysb, g.S, g.di);

            // 6) out = y @ out_w^T   (M x dm)
            dim3 g4((unsigned)((g.dm + 63) / 64), (unsigned)(Mrows / 128));
            gemm_f32_wmma<<<g4, 256, 0, stream>>>(
                ysb, g.di, g.out_w + (size_t)k * g.dm * g.di,
                outbuf, g.dm, Mrows, g.dm, g.di, nullptr, 0);
        }
    };

    // direction a: (B, 256, 512)
    run_stack(ga, x, buf0);

    // transpose (B,256,512) -> (B,512,256)
    {
        dim3 tg(512 / 32, 256 / 32, 2);
        transpose_k<<<tg, 256, 0, stream>>>(buf0, buf1, 256, 512);
    }

    // direction b: (B, 512, 256), in-place ping on buf1 (stream-serialized)
    run_stack(gb, buf1, buf1);

    // out = x + transpose(h)
    {
        const int tot = 2 * 256 * 512;
        residual_k<<<(tot + 255) / 256, 256, 0, stream>>>(x, buf1, (float*)d_out, 2, 256, 512);
    }
}